// MCUDetectionLoss_58540404244617
// MI455X (gfx1250) — compile-verified
//
#include <hip/hip_runtime.h>

// MCU detection loss for MI455X (gfx1250).
// B=32, T=64, C=81 (1 obj + 80 classes); P4: 80x80, P5: 40x40.
#define BB   32
#define TT   64
#define CCH  81
#define NC   80

typedef __attribute__((ext_vector_type(2))) float v2f;
typedef __attribute__((ext_vector_type(8))) float v8f;

#if defined(__has_builtin)
#if __has_builtin(__builtin_amdgcn_wmma_f32_16x16x4_f32)
#define HAVE_WMMA_F32X4 1
#endif
#endif

__device__ __forceinline__ float softplus_f(float x) {
  // stable log(1 + e^x)
  return fmaxf(x, 0.0f) + log1pf(expf(-fabsf(x)));
}
__device__ __forceinline__ float sigmoid_f(float x) {
  return 1.0f / (1.0f + expf(-x));
}

__global__ __launch_bounds__(256) void zero_i32(int* __restrict__ p, int n) {
  int i = blockIdx.x * 256 + threadIdx.x;
  if (i < n) p[i] = 0;
}

// One wave32 per (scale, b, t) target.  2*B*T = 4096 waves, block = 4 waves.
__global__ __launch_bounds__(128)
void target_loss_kernel(const float* __restrict__ cls4, const float* __restrict__ reg4,
                        const float* __restrict__ cls5, const float* __restrict__ reg5,
                        const int* __restrict__ c4, const float* __restrict__ b4, const float* __restrict__ m4,
                        const int* __restrict__ c5, const float* __restrict__ b5, const float* __restrict__ m5,
                        int* __restrict__ flags4, int* __restrict__ flags5,
                        float4* __restrict__ tpart) {
  int gtid = blockIdx.x * 128 + threadIdx.x;
  int wid  = gtid >> 5;        // 0..4095
  int lane = gtid & 31;
  int s    = wid >> 11;        // 0 = P4, 1 = P5   (wave-uniform)
  int idx  = wid & 2047;
  int b    = idx / TT;
  int t    = idx - b * TT;

  const float* cls_p = s ? cls5 : cls4;
  const float* reg_p = s ? reg5 : reg4;
  const int*   tcls  = s ? c5 : c4;
  const float* tbox  = s ? b5 : b4;
  const float* tmask = s ? m5 : m4;
  int* flags = s ? flags5 : flags4;
  const int W  = s ? 40 : 80;
  const int H  = W;
  const int HW = W * H;

  int bt = b * TT + t;
  float mask = tmask[bt];
  float bx = tbox[bt * 4 + 0], by = tbox[bt * 4 + 1];
  float bw = tbox[bt * 4 + 2], bh = tbox[bt * 4 + 3];
  float tx = bx * (float)W, ty = by * (float)H;
  float tw = bw * (float)W, th = bh * (float)H;
  int gx = (int)fminf(fmaxf(tx, 0.0f), (float)(W - 1));
  int gy = (int)fminf(fmaxf(ty, 0.0f), (float)(H - 1));
  int cell = gy * W + gx;

  if (lane == 0 && mask > 0.0f) flags[b * HW + cell] = 1;  // obj_map mark (idempotent)

  // --- bbox smooth-L1 (all lanes redundantly; wave-uniform result) ---
  size_t rbase = (size_t)b * 4 * HW + cell;
  float r0 = reg_p[rbase + 0 * (size_t)HW];
  float r1 = reg_p[rbase + 1 * (size_t)HW];
  float r2 = reg_p[rbase + 2 * (size_t)HW];
  float r3 = reg_p[rbase + 3 * (size_t)HW];
  float dx = sigmoid_f(r0), dy = sigmoid_f(r1);
  float dw = expf(fminf(fmaxf(r2, -4.0f), 4.0f));
  float dh = expf(fminf(fmaxf(r3, -4.0f), 4.0f));
  float px = (float)gx + dx, py = (float)gy + dy;
  float pb0 = px - 0.5f * dw, pb1 = py - 0.5f * dh;
  float pb2 = px + 0.5f * dw, pb3 = py + 0.5f * dh;
  float tb0 = tx - 0.5f * tw, tb1 = ty - 0.5f * th;
  float tb2 = tx + 0.5f * tw, tb3 = ty + 0.5f * th;
  float sl1 = 0.0f, d;
  d = pb0 - tb0; sl1 += (fabsf(d) < 1.0f) ? 0.5f * d * d : fabsf(d) - 0.5f;
  d = pb1 - tb1; sl1 += (fabsf(d) < 1.0f) ? 0.5f * d * d : fabsf(d) - 0.5f;
  d = pb2 - tb2; sl1 += (fabsf(d) < 1.0f) ? 0.5f * d * d : fabsf(d) - 0.5f;
  d = pb3 - tb3; sl1 += (fabsf(d) < 1.0f) ? 0.5f * d * d : fabsf(d) - 0.5f;
  float lb = (sl1 * 0.25f) * mask;

  // --- objectness ---
  size_t cbase = (size_t)b * CCH * HW + cell;
  float obj = cls_p[cbase];
  float lo = softplus_f(-obj) * mask;

  // --- focal classification over 80 classes, strided across the wave ---
  int cid = tcls[bt];
  float fsum = 0.0f;
  for (int c = lane; c < NC; c += 32) {
    float x  = cls_p[cbase + (size_t)(1 + c) * HW];
    float tt2 = (c == cid) ? 1.0f : 0.0f;
    float bce = softplus_f(x) - x * tt2;           // logaddexp(0,x) - x*t
    float p   = sigmoid_f(x);
    float pt  = p * tt2 + (1.0f - p) * (1.0f - tt2);
    float om  = 1.0f - pt;
    fsum += 0.25f * om * om * bce;                 // ALPHA=0.25, GAMMA=2
  }
  for (int o = 16; o > 0; o >>= 1) fsum += __shfl_xor(fsum, o, 32);
  float lc = (fsum * (1.0f / (float)NC)) * mask;

  if (lane == 0) tpart[wid] = make_float4(lb, lo, lc, mask);
}

// Background BCE over channel 0; one thread per cell; per-block partials.
__global__ __launch_bounds__(256)
void bg_kernel(const float* __restrict__ cls_p, const int* __restrict__ flags,
               int HW, float2* __restrict__ part) {
  int i  = blockIdx.x * 256 + threadIdx.x;   // grid sized exactly: B*HW / 256
  int b  = i / HW;
  int hw = i - b * HW;
  float x = cls_p[(size_t)b * CCH * HW + hw];
  int f = flags[i];
  float s = (f == 0) ? softplus_f(x) : 0.0f;
  float c = (f == 0) ? 1.0f : 0.0f;
  for (int o = 16; o > 0; o >>= 1) { s += __shfl_xor(s, o, 32); c += __shfl_xor(c, o, 32); }
  __shared__ float ss[8], sc[8];
  int wave = threadIdx.x >> 5, lane = threadIdx.x & 31;
  if (lane == 0) { ss[wave] = s; sc[wave] = c; }
  __syncthreads();
  if (wave == 0) {
    s = (lane < 8) ? ss[lane] : 0.0f;
    c = (lane < 8) ? sc[lane] : 0.0f;
    for (int o = 4; o > 0; o >>= 1) { s += __shfl_xor(s, o, 32); c += __shfl_xor(c, o, 32); }
    if (lane == 0) part[blockIdx.x] = make_float2(s, c);
  }
}

// Single wave32: WMMA (16x16x4 f32, A = ones) deterministic sum of 4096
// target partials, then combine with background partials -> scalar loss.
__global__ __launch_bounds__(32)
void final_kernel(const float4* __restrict__ tpart,
                  const float2* __restrict__ bg4, const float2* __restrict__ bg5,
                  float* __restrict__ out) {
  int lane = threadIdx.x;
  float tlb, tlo, tlc, tn;
#ifdef HAVE_WMMA_F32X4
  v2f ones = {1.0f, 1.0f};
  v8f alb = {}, alo = {}, alc = {}, an = {};
  for (int base = 0; base < 4096; base += 64) {           // uniform; EXEC all-ones
    float4 p0 = tpart[base + 2 * lane];
    float4 p1 = tpart[base + 2 * lane + 1];
    v2f blb = {p0.x, p1.x};
    v2f blo = {p0.y, p1.y};
    v2f blc = {p0.z, p1.z};
    v2f bn  = {p0.w, p1.w};
    alb = __builtin_amdgcn_wmma_f32_16x16x4_f32(false, ones, false, blb, (short)0, alb, false, false);
    alo = __builtin_amdgcn_wmma_f32_16x16x4_f32(false, ones, false, blo, (short)0, alo, false, false);
    alc = __builtin_amdgcn_wmma_f32_16x16x4_f32(false, ones, false, blc, (short)0, alc, false, false);
    an  = __builtin_amdgcn_wmma_f32_16x16x4_f32(false, ones, false, bn,  (short)0, an,  false, false);
  }
  // C/D VGPR0: lanes 0-15 = row M=0, lanes 16-31 = row M=8; both rows hold the
  // same column sums -> summing VGPR0 over all 32 lanes gives 2x the total.
  float vlb = alb[0], vlo = alo[0], vlc = alc[0], vn = an[0];
  for (int o = 16; o > 0; o >>= 1) {
    vlb += __shfl_xor(vlb, o, 32); vlo += __shfl_xor(vlo, o, 32);
    vlc += __shfl_xor(vlc, o, 32); vn  += __shfl_xor(vn,  o, 32);
  }
  tlb = vlb * 0.5f; tlo = vlo * 0.5f; tlc = vlc * 0.5f; tn = vn * 0.5f;
#else
  float vlb = 0.f, vlo = 0.f, vlc = 0.f, vn = 0.f;
  for (int i = lane; i < 4096; i += 32) {
    float4 p = tpart[i];
    vlb += p.x; vlo += p.y; vlc += p.z; vn += p.w;
  }
  for (int o = 16; o > 0; o >>= 1) {
    vlb += __shfl_xor(vlb, o, 32); vlo += __shfl_xor(vlo, o, 32);
    vlc += __shfl_xor(vlc, o, 32); vn  += __shfl_xor(vn,  o, 32);
  }
  tlb = vlb; tlo = vlo; tlc = vlc; tn = vn;
#endif
  float s4 = 0.f, c4 = 0.f, s5 = 0.f, c5 = 0.f;
  for (int i = lane; i < 800; i += 32) { float2 v = bg4[i]; s4 += v.x; c4 += v.y; }
  for (int i = lane; i < 200; i += 32) { float2 v = bg5[i]; s5 += v.x; c5 += v.y; }
  for (int o = 16; o > 0; o >>= 1) {
    s4 += __shfl_xor(s4, o, 32); c4 += __shfl_xor(c4, o, 32);
    s5 += __shfl_xor(s5, o, 32); c5 += __shfl_xor(c5, o, 32);
  }
  if (lane == 0) {
    float lo = tlo;
    lo += 0.05f * ((c4 > 0.0f) ? (s4 / fmaxf(c4, 1.0f)) : 0.0f);
    lo += 0.05f * ((c5 > 0.0f) ? (s5 / fmaxf(c5, 1.0f)) : 0.0f);
    float lb = (tn > 0.0f) ? (tlb / fmaxf(tn, 1.0f)) : tlb;
    float lc = (tn > 0.0f) ? (tlc / fmaxf(tn, 1.0f)) : tlc;
    lo = lo / fmaxf(tn, 1.0f);
    out[0] = 2.0f * lb + 1.0f * lo + 0.5f * lc;   // BBOX_W, OBJ_W, CLS_W
  }
}

extern "C" void kernel_launch(void* const* d_in, const int* in_sizes, int n_in,
                              void* d_out, int out_size, void* d_ws, size_t ws_size,
                              hipStream_t stream) {
  (void)in_sizes; (void)n_in; (void)out_size; (void)ws_size;
  const float* cls4 = (const float*)d_in[0];
  const float* reg4 = (const float*)d_in[1];
  const float* cls5 = (const float*)d_in[2];
  const float* reg5 = (const float*)d_in[3];
  const int*   c4   = (const int*)d_in[4];
  const float* b4   = (const float*)d_in[5];
  const float* m4   = (const float*)d_in[6];
  const int*   c5   = (const int*)d_in[7];
  const float* b5   = (const float*)d_in[8];
  const float* m5   = (const float*)d_in[9];

  // workspace layout
  const int FL4 = BB * 6400;   // 204800
  const int FL5 = BB * 1600;   //  51200
  int*    flags4 = (int*)d_ws;
  int*    flags5 = flags4 + FL4;
  float4* tpart  = (float4*)(flags5 + FL5);   // 4096 float4 (16B aligned offset)
  float2* bg4p   = (float2*)(tpart + 4096);   // 800 blocks
  float2* bg5p   = bg4p + 800;                // 200 blocks

  zero_i32<<<(FL4 + FL5 + 255) / 256, 256, 0, stream>>>(flags4, FL4 + FL5);
  target_loss_kernel<<<1024, 128, 0, stream>>>(cls4, reg4, cls5, reg5,
                                               c4, b4, m4, c5, b5, m5,
                                               flags4, flags5, tpart);
  bg_kernel<<<FL4 / 256, 256, 0, stream>>>(cls4, flags4, 6400, bg4p);
  bg_kernel<<<FL5 / 256, 256, 0, stream>>>(cls5, flags5, 1600, bg5p);
  final_kernel<<<1, 32, 0, stream>>>(tpart, bg4p, bg5p, (float*)d_out);
}